// SSM_36550171689114
// MI455X (gfx1250) — compile-verified
//
#include <hip/hip_runtime.h>
#include <hip/hip_bf16.h>
#include <stdint.h>

// Problem dims (match reference)
#define B_   2
#define HIN_ 64
#define D_   512
#define N_   64
#define L_   2048

typedef __attribute__((ext_vector_type(2))) float v2f;
typedef __attribute__((ext_vector_type(8))) float v8f;

// ---------------------------------------------------------------------------
// Batched GEMM via V_WMMA_F32_16X16X4_F32 with 4-way accumulator blocking:
//   Y[b, m, l] = sum_k W[m, k] * X[b, k, l]
// Each wave computes a 16(M) x 64(L) strip: the A fragment (from W) is loaded
// once per k-step and reused by 4 WMMAs (4x less A traffic + address math).
// grid = (L/512, M/16, Bt), block = 256 (8 wave32s; wave strip = 64 l's)
// ---------------------------------------------------------------------------
__global__ __launch_bounds__(256) void ssm_gemm_wmma_f32(
    const float* __restrict__ W, const float* __restrict__ X,
    float* __restrict__ Y, int M, int K, int Ldim) {
  const int wave = threadIdx.x >> 5;
  const int lane = threadIdx.x & 31;
  const int m0 = blockIdx.y << 4;
  const int l0 = ((blockIdx.x << 3) + wave) << 6;   // 64-wide strip per wave
  const int b  = blockIdx.z;

  const float* __restrict__ Xb = X + (size_t)b * K * Ldim;
  float* __restrict__ Yb = Y + (size_t)b * M * Ldim;

  const int mrow = lane & 15;          // A-frag row / B-frag col / D col
  const int p    = lane >> 4;          // K-pair selector (ISA 16x4 f32 layout)

  v8f c0 = {}, c1 = {}, c2 = {}, c3 = {};
  const float* __restrict__ wp = W + (size_t)(m0 + mrow) * K + 2 * p;
  const float* __restrict__ xp = Xb + (size_t)(2 * p) * Ldim + l0 + mrow;

  for (int k0 = 0; k0 < K; k0 += 4) {
    // A fragment: lane holds W[m, k0+2p] / W[m, k0+2p+1] -- shared by 4 WMMAs
    const float2 wv = *reinterpret_cast<const float2*>(wp);
    wp += 4;
    v2f a; a.x = wv.x; a.y = wv.y;
    // B fragments for the 4 l-tiles: rows k0+2p and k0+2p+1
    v2f b0, b1, b2, b3;
    b0.x = xp[0];  b0.y = xp[Ldim];
    b1.x = xp[16]; b1.y = xp[Ldim + 16];
    b2.x = xp[32]; b2.y = xp[Ldim + 32];
    b3.x = xp[48]; b3.y = xp[Ldim + 48];
    xp += (size_t)4 * Ldim;
    c0 = __builtin_amdgcn_wmma_f32_16x16x4_f32(false, a, false, b0, (short)0, c0, false, false);
    c1 = __builtin_amdgcn_wmma_f32_16x16x4_f32(false, a, false, b1, (short)0, c1, false, false);
    c2 = __builtin_amdgcn_wmma_f32_16x16x4_f32(false, a, false, b2, (short)0, c2, false, false);
    c3 = __builtin_amdgcn_wmma_f32_16x16x4_f32(false, a, false, b3, (short)0, c3, false, false);
  }
  // D layout: VGPR r -> (m = r + 8p, n = lane&15)
  float* __restrict__ yp = Yb + (size_t)(m0 + 8 * p) * Ldim + l0 + mrow;
#pragma unroll
  for (int r = 0; r < 8; ++r) {
    const size_t ro = (size_t)r * Ldim;
    yp[ro]      = c0[r];
    yp[ro + 16] = c1[r];
    yp[ro + 32] = c2[r];
    yp[ro + 48] = c3[r];
  }
}

// ---------------------------------------------------------------------------
// Selective scan, 1 recurrence state per thread; Bu/Cu streams staged into
// LDS with double-buffered GLOBAL_LOAD_ASYNC_TO_LDS_B128 (ASYNCcnt-tracked),
// so the 2048-step serial FMA chain overlaps the next chunk's data movement.
//   h_l = (delta*A[n]) * h_{l-1} + delta*Bu[n]*x
//   pooled[b,d] = (sum_l sum_n Cu[n]*h + (sum_l x)*Dvec[d]) / L
// grid = B * (D/4); block = 256 = 4 d-lanes x 64 n-lanes.
// ---------------------------------------------------------------------------
#define CL_   64            // l-chunk staged per buffer
#define ROWP_ 68            // padded LDS row stride (floats) -> bank spread
#define NCHUNK_ (L_ / CL_)

__global__ __launch_bounds__(256) void ssm_scan(
    const float* __restrict__ x, const float* __restrict__ delta,
    const float* __restrict__ Bu, const float* __restrict__ Cu,
    const float* __restrict__ A, const float* __restrict__ Dvec,
    float* __restrict__ pooled) {
  __shared__ __align__(16) float sBu[2][N_ * ROWP_];
  __shared__ __align__(16) float sCu[2][N_ * ROWP_];

  const int b  = blockIdx.x / (D_ / 4);
  const int dt = blockIdx.x % (D_ / 4);
  const int t  = threadIdx.x;
  const int dl = t & 3;                 // d within 4-wide tile (= copy quarter)
  const int n  = t >> 2;                // state index 0..63     (= copy row)
  const int d  = (dt << 2) + dl;

  const size_t xoff = ((size_t)b * D_ + d) * L_;
  const float* __restrict__ xr = x + xoff;
  const float* __restrict__ dr = delta + xoff;
  const size_t noff = ((size_t)b * N_ + n) * L_ + dl * 16;  // copy: row n, quarter dl
  const float* __restrict__ gbu = Bu + noff;
  const float* __restrict__ gcu = Cu + noff;
  const unsigned lbuBase = (unsigned)(uintptr_t)&sBu[0][n * ROWP_ + dl * 16];
  const unsigned lcuBase = (unsigned)(uintptr_t)&sCu[0][n * ROWP_ + dl * 16];
  const unsigned bufStride = (unsigned)(N_ * ROWP_ * sizeof(float));

  // Async-stage one 64-step chunk of Bu/Cu (this thread copies 16+16 floats).
  auto stage_chunk = [&](int c, int buf) {
    const float* gB = gbu + (size_t)c * CL_;
    const float* gC = gcu + (size_t)c * CL_;
    const unsigned lB = lbuBase + (unsigned)buf * bufStride;
    const unsigned lC = lcuBase + (unsigned)buf * bufStride;
#pragma unroll
    for (int j = 0; j < 4; ++j) {
      asm volatile("global_load_async_to_lds_b128 %0, %1, off"
                   :: "v"(lB + 16u * j), "v"(gB + 4 * j) : "memory");
      asm volatile("global_load_async_to_lds_b128 %0, %1, off"
                   :: "v"(lC + 16u * j), "v"(gC + 4 * j) : "memory");
    }
  };

  const float An = A[n];
  float h = 0.f, accy = 0.f, accx = 0.f;

  stage_chunk(0, 0);
  for (int c = 0; c < NCHUNK_; ++c) {
    const int buf = c & 1;
    if (c + 1 < NCHUNK_) {
      stage_chunk(c + 1, buf ^ 1);
      // next x/delta chunk -> global_prefetch_b8
      __builtin_prefetch(xr + (c + 1) * CL_, 0, 0);
      __builtin_prefetch(dr + (c + 1) * CL_, 0, 0);
      // 8 outstanding = the chunk just issued; chunk c is complete.
      asm volatile("s_wait_asynccnt 0x8" ::: "memory");
    } else {
      asm volatile("s_wait_asynccnt 0x0" ::: "memory");
    }
    __syncthreads();

    const float* __restrict__ bu_l = &sBu[buf][n * ROWP_];
    const float* __restrict__ cu_l = &sCu[buf][n * ROWP_];
    const int l0 = c * CL_;
    for (int lc = 0; lc < CL_; lc += 4) {
      const float4 xv = *reinterpret_cast<const float4*>(xr + l0 + lc);
      const float4 dv = *reinterpret_cast<const float4*>(dr + l0 + lc);
      const float4 bv = *reinterpret_cast<const float4*>(bu_l + lc);
      const float4 cv = *reinterpret_cast<const float4*>(cu_l + lc);
      const float xa[4] = {xv.x, xv.y, xv.z, xv.w};
      const float da[4] = {dv.x, dv.y, dv.z, dv.w};
      const float ba[4] = {bv.x, bv.y, bv.z, bv.w};
      const float ca[4] = {cv.x, cv.y, cv.z, cv.w};
#pragma unroll
      for (int s = 0; s < 4; ++s) {
        const float dd = da[s];
        const float dx = dd * xa[s];
        h = fmaf(dd * An, h, dx * ba[s]);
        accy = fmaf(ca[s], h, accy);
        accx += xa[s];
      }
    }
    __syncthreads();   // safe to overwrite this buffer at iteration c+2
  }

  // Reduce accy over n (64 groups) per d; one reduction for the whole scan.
  __shared__ float red[256];
  __shared__ float xs[4];
  red[t] = accy;
  if (n == 0) xs[dl] = accx;
  __syncthreads();
  if (t < 4) {
    float s = 0.f;
#pragma unroll
    for (int g = 0; g < 64; ++g) s += red[(g << 2) + t];
    const int dg = (dt << 2) + t;
    pooled[(size_t)b * D_ + dg] = (s + xs[t] * Dvec[dg]) * (1.0f / (float)L_);
  }
}

// ---------------------------------------------------------------------------
// Readout: out[b,i] = sum_d W_down[i,d] * pooled[b,d]
// ---------------------------------------------------------------------------
__global__ __launch_bounds__(256) void ssm_readout(
    const float* __restrict__ W_down, const float* __restrict__ pooled,
    float* __restrict__ out) {
  const int o = blockIdx.x;
  const int b = o / 10;
  const int i = o % 10;
  float s = 0.f;
  for (int d = threadIdx.x; d < D_; d += 256)
    s += W_down[(size_t)i * D_ + d] * pooled[(size_t)b * D_ + d];
  __shared__ float red[256];
  red[threadIdx.x] = s;
  __syncthreads();
  for (int w = 128; w > 0; w >>= 1) {
    if (threadIdx.x < w) red[threadIdx.x] += red[threadIdx.x + w];
    __syncthreads();
  }
  if (threadIdx.x == 0) out[o] = red[0];
}

// ---------------------------------------------------------------------------
extern "C" void kernel_launch(void* const* d_in, const int* in_sizes, int n_in,
                              void* d_out, int out_size, void* d_ws, size_t ws_size,
                              hipStream_t stream) {
  const float* u      = (const float*)d_in[0];  // (B, HIN, L)
  const float* W_up   = (const float*)d_in[1];  // (D, HIN)
  const float* A      = (const float*)d_in[2];  // (N)
  const float* Bmat   = (const float*)d_in[3];  // (N, D)
  const float* Cmat   = (const float*)d_in[4];  // (N, D)
  const float* Dvec   = (const float*)d_in[5];  // (D)
  const float* Delta  = (const float*)d_in[6];  // (D, D)
  const float* W_down = (const float*)d_in[7];  // (10, D)
  float* out = (float*)d_out;                   // (B, 10) fp32

  // Workspace layout (floats)
  float* ws     = (float*)d_ws;
  float* x      = ws;                                    // B*D*L
  float* delta  = x     + (size_t)B_ * D_ * L_;          // B*D*L
  float* Bu     = delta + (size_t)B_ * D_ * L_;          // B*N*L
  float* Cu     = Bu    + (size_t)B_ * N_ * L_;          // B*N*L
  float* pooled = Cu    + (size_t)B_ * N_ * L_;          // B*D

  const dim3 blk(256);
  // x = W_up @ u          (M=512, K=64)
  ssm_gemm_wmma_f32<<<dim3(L_ / 512, D_ / 16, B_), blk, 0, stream>>>(
      W_up, u, x, D_, HIN_, L_);
  // delta = Delta @ x     (M=512, K=512)
  ssm_gemm_wmma_f32<<<dim3(L_ / 512, D_ / 16, B_), blk, 0, stream>>>(
      Delta, x, delta, D_, D_, L_);
  // Bu = Bmat @ x         (M=64, K=512)
  ssm_gemm_wmma_f32<<<dim3(L_ / 512, N_ / 16, B_), blk, 0, stream>>>(
      Bmat, x, Bu, N_, D_, L_);
  // Cu = Cmat @ x         (M=64, K=512)
  ssm_gemm_wmma_f32<<<dim3(L_ / 512, N_ / 16, B_), blk, 0, stream>>>(
      Cmat, x, Cu, N_, D_, L_);
  // selective scan -> pooled
  ssm_scan<<<dim3(B_ * (D_ / 4)), blk, 0, stream>>>(
      x, delta, Bu, Cu, A, Dvec, pooled);
  // readout -> out
  ssm_readout<<<dim3(B_ * 10), blk, 0, stream>>>(W_down, pooled, out);
}